// maTransformerBlock_59176059404461
// MI455X (gfx1250) — compile-verified
//
#include <hip/hip_runtime.h>
#include <hip/hip_bf16.h>

// ---------------------------------------------------------------------------
// Reference shapes: B=32768, L=103, conv filters width 8 -> 96 outputs,
// CHUNK=3 -> 32 chunks, cross-batch "stack/chunk" permutation, 32x32 attn,
// linear 96->2.  Memory bound (~337MB reads @ 23.3TB/s ~ 14.5us).
// ---------------------------------------------------------------------------

#define LSEQ   103
#define LOUT   96
#define FW     8
#define CD     4      // dna channels
#define CP     21     // prot channels
#define WPB    8      // waves (attention rows) per block in pass 2

typedef __attribute__((ext_vector_type(2))) float v2f;
typedef __attribute__((ext_vector_type(8))) float v8f;

// ---------------------------------------------------------------------------
// Pass 1: full-height valid conv + ReLU for dna & prot.
// One block per batch; rows staged in LDS, each thread t<96 produces d[b,t],
// p[b,t].  Weights staged in LDS once (tiny).
// ---------------------------------------------------------------------------
__global__ void __launch_bounds__(128)
conv_relu_kernel(const float* __restrict__ dna,
                 const float* __restrict__ prot,
                 const float* __restrict__ w_dna,
                 const float* __restrict__ w_prot,
                 float* __restrict__ dbuf,   // (B,96)
                 float* __restrict__ pbuf)   // (B,96)
{
    __shared__ float s_dna[CD * LSEQ];   // 412 f
    __shared__ float s_prot[CP * LSEQ];  // 2163 f
    __shared__ float s_wd[CD * FW];      // 32 f
    __shared__ float s_wp[CP * FW];      // 168 f

    const int b = blockIdx.x;
    const int t = threadIdx.x;
    const float* gd = dna  + (size_t)b * (CD * LSEQ);
    const float* gp = prot + (size_t)b * (CP * LSEQ);

    for (int i = t; i < CD * LSEQ; i += blockDim.x) s_dna[i]  = gd[i];
    for (int i = t; i < CP * LSEQ; i += blockDim.x) s_prot[i] = gp[i];
    for (int i = t; i < CD * FW;   i += blockDim.x) s_wd[i]   = w_dna[i];
    for (int i = t; i < CP * FW;   i += blockDim.x) s_wp[i]   = w_prot[i];
    __syncthreads();

    if (t < LOUT) {
        float accd = 0.f;
#pragma unroll
        for (int c = 0; c < CD; ++c)
#pragma unroll
            for (int f = 0; f < FW; ++f)
                accd = fmaf(s_dna[c * LSEQ + t + f], s_wd[c * FW + f], accd);

        float accp = 0.f;
        for (int c = 0; c < CP; ++c)
#pragma unroll
            for (int f = 0; f < FW; ++f)
                accp = fmaf(s_prot[c * LSEQ + t + f], s_wp[c * FW + f], accp);

        dbuf[(size_t)b * LOUT + t] = fmaxf(accd, 0.f);
        pbuf[(size_t)b * LOUT + t] = fmaxf(accp, 0.f);
    }
}

// ---------------------------------------------------------------------------
// Pass 2: permuted gather + attention (WMMA f32 16x16x4) + softmax + linear.
// One wave32 per output row b2.  Permutation algebra:
//   g = 32*b2 + lane  ->  sd[b2][lane][c] = d[g % B][3*(g / B) + c]
// ---------------------------------------------------------------------------
__global__ void __launch_bounds__(WPB * 32)
attn_linear_kernel(const float* __restrict__ dbuf,
                   const float* __restrict__ pbuf,
                   const float* __restrict__ w_lin,  // (2,96)
                   const float* __restrict__ b_lin,  // (2,)
                   float* __restrict__ out,          // (B,2)
                   int bsz)
{
    __shared__ float s_sd[WPB][32][4];    // K padded 3->4 (zeros)
    __shared__ float s_sp[WPB][32][4];    // pre-scaled by 1/sqrt(3)
    __shared__ float s_attn[WPB][32][33]; // 32x32, stride 33 (bank-conflict-free)
    __shared__ float s_out[WPB][32][3];

    const int lane = threadIdx.x & 31;
    const int wave = threadIdx.x >> 5;
    int b2 = blockIdx.x * WPB + wave;
    const bool valid = (b2 < bsz);
    if (!valid) b2 = 0;  // keep wave fully active; never stores

    const int half = lane >> 4;   // 0: K0/K1 half, 1: K2/K3 half
    const int l16  = lane & 15;

    // ---- permuted gather: one contiguous 3-float read per lane per tensor
    const unsigned g  = (unsigned)b2 * 32u + (unsigned)lane;
    const unsigned n  = g / (unsigned)bsz;       // chunk index 0..31
    const unsigned bs = g % (unsigned)bsz;       // source batch
    const size_t base = (size_t)bs * LOUT + (size_t)n * 3u;
    const float scale = 0.57735026918962576451f; // 1/sqrt(3)

    s_sd[wave][lane][0] = dbuf[base + 0];
    s_sd[wave][lane][1] = dbuf[base + 1];
    s_sd[wave][lane][2] = dbuf[base + 2];
    s_sd[wave][lane][3] = 0.f;
    s_sp[wave][lane][0] = pbuf[base + 0] * scale;
    s_sp[wave][lane][1] = pbuf[base + 1] * scale;
    s_sp[wave][lane][2] = pbuf[base + 2] * scale;
    s_sp[wave][lane][3] = 0.f;
    __syncthreads();

    // ---- logits = sp_scaled (32x3) @ sd^T (3x32) as 4 WMMA f32 16x16x4 tiles
    // A frag (16x4): lanes0-15 hold K0(v0)/K1(v1); lanes16-31 hold K2/K3.
    // B frag (4x16): per-column mirror of A.  K=3 column is the zero pad.
    v2f afr[2], bfr[2];
#pragma unroll
    for (int mi = 0; mi < 2; ++mi) {
        const int row = mi * 16 + l16;
        afr[mi].x = s_sp[wave][row][half * 2 + 0];
        afr[mi].y = s_sp[wave][row][half * 2 + 1];
    }
#pragma unroll
    for (int ni = 0; ni < 2; ++ni) {
        const int col = ni * 16 + l16;
        bfr[ni].x = s_sd[wave][col][half * 2 + 0];
        bfr[ni].y = s_sd[wave][col][half * 2 + 1];
    }
#pragma unroll
    for (int mi = 0; mi < 2; ++mi) {
#pragma unroll
        for (int ni = 0; ni < 2; ++ni) {
            v8f c = {0.f, 0.f, 0.f, 0.f, 0.f, 0.f, 0.f, 0.f};
            c = __builtin_amdgcn_wmma_f32_16x16x4_f32(
                    false, afr[mi], false, bfr[ni], (short)0, c, false, false);
            // C/D layout: vgpr r -> (M = r | 8+r depending on lane half, N = l16)
#pragma unroll
            for (int r = 0; r < 8; ++r)
                s_attn[wave][mi * 16 + half * 8 + r][ni * 16 + l16] = c[r];
        }
    }
    __syncthreads();

    // ---- row softmax: lane = row, full 32-wide row in registers
    {
        float rowv[32];
        float m = -3.402823466e38f;
#pragma unroll
        for (int k = 0; k < 32; ++k) {
            rowv[k] = s_attn[wave][lane][k];
            m = fmaxf(m, rowv[k]);
        }
        float sum = 0.f;
#pragma unroll
        for (int k = 0; k < 32; ++k) {
            rowv[k] = __expf(rowv[k] - m);
            sum += rowv[k];
        }
        const float inv = 1.0f / sum;
#pragma unroll
        for (int k = 0; k < 32; ++k)
            s_attn[wave][lane][k] = rowv[k] * inv;
    }
    __syncthreads();

    // ---- out = attn (32x32) @ sd (32x3):  2 tiles x 8 chained K-steps,
    //      N padded 3->16 with zero columns.
#pragma unroll
    for (int mi = 0; mi < 2; ++mi) {
        v8f acc = {0.f, 0.f, 0.f, 0.f, 0.f, 0.f, 0.f, 0.f};
        const int arow = mi * 16 + l16;
#pragma unroll
        for (int k0 = 0; k0 < 32; k0 += 4) {
            v2f a, b;
            a.x = s_attn[wave][arow][k0 + half * 2 + 0];
            a.y = s_attn[wave][arow][k0 + half * 2 + 1];
            b.x = (l16 < 3) ? s_sd[wave][k0 + half * 2 + 0][l16] : 0.f;
            b.y = (l16 < 3) ? s_sd[wave][k0 + half * 2 + 1][l16] : 0.f;
            acc = __builtin_amdgcn_wmma_f32_16x16x4_f32(
                      false, a, false, b, (short)0, acc, false, false);
        }
        if (l16 < 3) {
#pragma unroll
            for (int r = 0; r < 8; ++r)
                s_out[wave][mi * 16 + half * 8 + r][l16] = acc[r];
        }
    }
    __syncthreads();

    // ---- linear 96 -> 2: lane owns flattened chunk row `lane` (3 values),
    //      then wave32 shuffle-xor reduction.
    float o0 = 0.f, o1 = 0.f;
#pragma unroll
    for (int c = 0; c < 3; ++c) {
        const float v = s_out[wave][lane][c];
        const int j = lane * 3 + c;
        o0 = fmaf(v, w_lin[j], o0);
        o1 = fmaf(v, w_lin[LOUT + j], o1);
    }
#pragma unroll
    for (int off = 16; off > 0; off >>= 1) {
        o0 += __shfl_xor(o0, off, 32);
        o1 += __shfl_xor(o1, off, 32);
    }
    if (valid && lane == 0) {
        out[(size_t)b2 * 2 + 0] = o0 + b_lin[0];
        out[(size_t)b2 * 2 + 1] = o1 + b_lin[1];
    }
}

// ---------------------------------------------------------------------------
extern "C" void kernel_launch(void* const* d_in, const int* in_sizes, int n_in,
                              void* d_out, int out_size, void* d_ws, size_t ws_size,
                              hipStream_t stream) {
    const float* dna    = (const float*)d_in[0];
    const float* prot   = (const float*)d_in[1];
    const float* w_dna  = (const float*)d_in[2];
    const float* w_prot = (const float*)d_in[3];
    const float* w_lin  = (const float*)d_in[4];
    const float* b_lin  = (const float*)d_in[5];
    float* out = (float*)d_out;

    const int bsz = in_sizes[0] / (CD * LSEQ);   // 32768

    float* dbuf = (float*)d_ws;                  // (B,96)
    float* pbuf = dbuf + (size_t)bsz * LOUT;     // (B,96)

    conv_relu_kernel<<<bsz, 128, 0, stream>>>(dna, prot, w_dna, w_prot,
                                              dbuf, pbuf);

    const int blocks = (bsz + WPB - 1) / WPB;
    attn_linear_kernel<<<blocks, WPB * 32, 0, stream>>>(dbuf, pbuf, w_lin,
                                                        b_lin, out, bsz);
}